// VarLin_Layer_1571958030970
// MI455X (gfx1250) — compile-verified
//
#include <hip/hip_runtime.h>

typedef __attribute__((ext_vector_type(2))) float v2f;
typedef __attribute__((ext_vector_type(8))) float v8f;

constexpr int V_DIM = 16;
constexpr int B_DIM = 4;
constexpr int T_DIM = 4096;
constexpr int D = 64;                          // D_IN == D_OUT
constexpr int ROWS_PER_V = B_DIM * T_DIM;      // 16384
constexpr int MT_PER_V   = ROWS_PER_V / 16;    // 1024 M-tiles per variable
constexpr int WAVES_PER_WG = 8;                // 256 threads, wave32
constexpr int WAVE_MTILES  = 4;                // M-tiles per wave (amortize W regs)
constexpr int MT_PER_WG    = WAVES_PER_WG * WAVE_MTILES;  // 32
constexpr int WGS_PER_V    = MT_PER_V / MT_PER_WG;        // 32

__global__ __launch_bounds__(256)
void varlin_wmma_f32(const float* __restrict__ x,
                     const long long* __restrict__ var_idx,
                     const float* __restrict__ w,
                     float* __restrict__ out) {
  const int wg   = blockIdx.x;
  const int v    = wg / WGS_PER_V;           // uniform per workgroup
  const int blk  = wg % WGS_PER_V;
  const int wave = threadIdx.x >> 5;
  const int lane = threadIdx.x & 31;
  const int lhi  = lane >> 4;                // lane half: 0 or 1
  const int llo  = lane & 15;

  const int widx = (int)var_idx[v];          // wave-uniform gather
  const float* __restrict__ W = w + (size_t)widx * (D * D);

  // --- Load full 64x64 weight matrix into per-wave B fragments (once). ---
  // B fragment for V_WMMA_F32_16X16X4_F32 is 4x16 (KxN) f32 in 2 VGPRs:
  //   VGPR0: lanes0-15 -> K=0, lanes16-31 -> K=2 ; VGPR1: K=1 / K=3 ; N = lane&15
  v2f bfrag[16][4];
  #pragma unroll
  for (int kk = 0; kk < 16; ++kk) {
    const int kbase = kk * 4 + lhi * 2;
    #pragma unroll
    for (int nt = 0; nt < 4; ++nt) {
      const int n = nt * 16 + llo;
      v2f bf;
      bf.x = W[(kbase + 0) * D + n];
      bf.y = W[(kbase + 1) * D + n];
      bfrag[kk][nt] = bf;
    }
  }

  const size_t vrow = (size_t)v * ROWS_PER_V;

  for (int i = 0; i < WAVE_MTILES; ++i) {
    const int mt = blk * MT_PER_WG + wave * WAVE_MTILES + i;
    const size_t row0 = vrow + (size_t)mt * 16;

    // --- A fragments: 16x4 (MxK) f32 per chunk; one global_load_b64/lane. ---
    //   lanes0-15: row = llo, K pair (k, k+1); lanes16-31: K pair (k+2, k+3)
    const float* __restrict__ xr = x + (row0 + (size_t)llo) * D + lhi * 2;
    v2f afrag[16];
    #pragma unroll
    for (int kk = 0; kk < 16; ++kk)
      afrag[kk] = *reinterpret_cast<const v2f*>(xr + kk * 4);

    v8f acc0 = {}, acc1 = {}, acc2 = {}, acc3 = {};
    #pragma unroll
    for (int kk = 0; kk < 16; ++kk) {
      acc0 = __builtin_amdgcn_wmma_f32_16x16x4_f32(false, afrag[kk], false, bfrag[kk][0], (short)0, acc0, false, false);
      acc1 = __builtin_amdgcn_wmma_f32_16x16x4_f32(false, afrag[kk], false, bfrag[kk][1], (short)0, acc1, false, false);
      acc2 = __builtin_amdgcn_wmma_f32_16x16x4_f32(false, afrag[kk], false, bfrag[kk][2], (short)0, acc2, false, false);
      acc3 = __builtin_amdgcn_wmma_f32_16x16x4_f32(false, afrag[kk], false, bfrag[kk][3], (short)0, acc3, false, false);
    }

    // --- Store D tiles: VGPR r -> M = r (lanes0-15) / r+8 (lanes16-31). ---
    float* __restrict__ orow = out + (row0 + (size_t)lhi * 8) * D + llo;
    #pragma unroll
    for (int r = 0; r < 8; ++r) {
      orow[(size_t)r * D +  0] = acc0[r];
      orow[(size_t)r * D + 16] = acc1[r];
      orow[(size_t)r * D + 32] = acc2[r];
      orow[(size_t)r * D + 48] = acc3[r];
    }
  }
}

extern "C" void kernel_launch(void* const* d_in, const int* in_sizes, int n_in,
                              void* d_out, int out_size, void* d_ws, size_t ws_size,
                              hipStream_t stream) {
  const float*      x       = (const float*)d_in[0];
  const long long*  var_idx = (const long long*)d_in[1];
  const float*      w       = (const float*)d_in[2];
  float*            out     = (float*)d_out;

  dim3 grid(V_DIM * WGS_PER_V);   // 512 workgroups
  dim3 block(256);                // 8 waves (wave32)
  hipLaunchKernelGGL(varlin_wmma_f32, grid, block, 0, stream, x, var_idx, w, out);
}